// RPN_3D_loss_20787641712668
// MI455X (gfx1250) — compile-verified
//
#include <hip/hip_runtime.h>

#define BB 4
#define NN 126720
#define GG 32
#define CC 4
#define KK 1024
#define NBLK_A (NN / 256)          // 495
#define NPART (NBLK_A * BB)        // 1980

__device__ __constant__ float kEPS = 1e-6f;
#define FG_T 0.5f
#define NMS_T 0.4f

typedef __attribute__((ext_vector_type(2))) float v2f;
typedef __attribute__((ext_vector_type(8))) float v8f;

// ---------- helpers ----------
__device__ inline float sl1(float x) {
    float ax = fabsf(x);
    return ax < 1.f ? 0.5f * ax * ax : ax - 1.f;
}

__device__ inline float ioup(float4 p, float4 q) {
    float areaA = fmaxf(p.z - p.x, 0.f) * fmaxf(p.w - p.y, 0.f);
    float areaB = fmaxf(q.z - q.x, 0.f) * fmaxf(q.w - q.y, 0.f);
    float iw = fmaxf(fminf(p.z, q.z) - fmaxf(p.x, q.x), 0.f);
    float ih = fmaxf(fminf(p.w, q.w) - fmaxf(p.y, q.y), 0.f);
    float inter = iw * ih;
    return inter / (areaA + areaB - inter + 1e-6f);
}

__device__ inline float lval(float4 p, float4 q) {   // pruned IoU (P matrix entry)
    float v = ioup(p, q);
    return v > NMS_T ? v : 0.f;
}

__device__ inline unsigned long long make_key(float f, int n) {
    unsigned int u = __float_as_uint(f);
    u = (u & 0x80000000u) ? ~u : (u | 0x80000000u);   // total order, ascending
    // embed inverted index: equal scores -> smaller n is larger key (jax first-index tie-break)
    return ((unsigned long long)u << 32) | (unsigned long long)(0xFFFFFFFFu - (unsigned int)n);
}

// ---------- kernel 1: per-anchor assignment + losses + scores ----------
__global__ __launch_bounds__(256) void k_anchor(
    const float* __restrict__ cls, const float* __restrict__ prob,
    const float* __restrict__ b2d, const float* __restrict__ b3d,
    const float* __restrict__ rois, const float* __restrict__ gtb,
    const int* __restrict__ gtl, const float* __restrict__ gt3,
    const float* __restrict__ means, const float* __restrict__ stds,
    float* __restrict__ scores, float4* __restrict__ partials)
{
    const int tid = threadIdx.x;
    const int b = blockIdx.y;
    const int n = blockIdx.x * 256 + tid;

    __shared__ float4 sgt[GG];
    __shared__ int slab[GG];
    __shared__ float sg3[GG * 7];
    __shared__ float sm[11], ss[11];
    if (tid < GG) { sgt[tid] = ((const float4*)gtb)[b * GG + tid]; slab[tid] = gtl[b * GG + tid]; }
    for (int i = tid; i < GG * 7; i += 256) sg3[i] = gt3[b * GG * 7 + i];
    if (tid < 11) { sm[tid] = means[tid]; ss[tid] = stds[tid]; }
    __syncthreads();

    float cAcc = 0.f, l2Acc = 0.f, l3Acc = 0.f, fgAcc = 0.f;
    if (n < NN) {
        float4 r = ((const float4*)rois)[n];
        float aw = r.z - r.x, ah = r.w - r.y;
        float areaA = fmaxf(aw, 0.f) * fmaxf(ah, 0.f);
        float best = -1.f; int bi = 0;
        #pragma unroll 4
        for (int g = 0; g < GG; ++g) {
            float4 q = sgt[g];
            float areaB = fmaxf(q.z - q.x, 0.f) * fmaxf(q.w - q.y, 0.f);
            float iw = fmaxf(fminf(r.z, q.z) - fmaxf(r.x, q.x), 0.f);
            float ih = fmaxf(fminf(r.w, q.w) - fmaxf(r.y, q.y), 0.f);
            float inter = iw * ih;
            float iou = inter / (areaA + areaB - inter + 1e-6f);
            if (iou > best) { best = iou; bi = g; }
        }
        float fg = best >= FG_T ? 1.f : 0.f;
        int label = (best >= FG_T) ? slab[bi] : 0;

        size_t base = ((size_t)b * NN + n) * CC;
        float x0 = cls[base], x1 = cls[base + 1], x2 = cls[base + 2], x3 = cls[base + 3];
        float mx = fmaxf(fmaxf(x0, x1), fmaxf(x2, x3));
        float lse = mx + logf(expf(x0 - mx) + expf(x1 - mx) + expf(x2 - mx) + expf(x3 - mx));
        float xl = label == 0 ? x0 : (label == 1 ? x1 : (label == 2 ? x2 : x3));
        cAcc = lse - xl;   // -log_softmax[label]

        // 2D regression target
        float4 q = sgt[bi];
        float acx = r.x + 0.5f * aw, acy = r.y + 0.5f * ah;
        float gw = q.z - q.x, gh = q.w - q.y;
        float gcx = q.x + 0.5f * gw, gcy = q.y + 0.5f * gh;
        float t0 = (gcx - acx) / (aw + 1e-6f), t1 = (gcy - acy) / (ah + 1e-6f);
        float t2 = logf(gw / (aw + 1e-6f) + 1e-6f), t3 = logf(gh / (ah + 1e-6f) + 1e-6f);
        t0 = (t0 - sm[0]) / ss[0]; t1 = (t1 - sm[1]) / ss[1];
        t2 = (t2 - sm[2]) / ss[2]; t3 = (t3 - sm[3]) / ss[3];
        size_t b4 = ((size_t)b * NN + n) * 4;
        l2Acc = (sl1(b2d[b4] - t0) + sl1(b2d[b4 + 1] - t1) +
                 sl1(b2d[b4 + 2] - t2) + sl1(b2d[b4 + 3] - t3)) * fg;

        // 3D regression target
        size_t b7 = ((size_t)b * NN + n) * 7;
        float s3 = 0.f;
        #pragma unroll
        for (int j = 0; j < 7; ++j) {
            float t = (sg3[bi * 7 + j] - sm[4 + j]) / ss[4 + j];
            s3 += sl1(b3d[b7 + j] - t);
        }
        l3Acc = s3 * fg;
        fgAcc = fg;

        scores[(size_t)b * NN + n] = fmaxf(fmaxf(prob[base + 1], prob[base + 2]), prob[base + 3]);
    }

    __shared__ float4 red[256];
    red[tid] = make_float4(cAcc, l2Acc, l3Acc, fgAcc);
    __syncthreads();
    for (int s = 128; s > 0; s >>= 1) {
        if (tid < s) {
            float4 a = red[tid], c = red[tid + s];
            red[tid] = make_float4(a.x + c.x, a.y + c.y, a.z + c.z, a.w + c.w);
        }
        __syncthreads();
    }
    if (tid == 0) partials[blockIdx.y * gridDim.x + blockIdx.x] = red[0];
}

// ---------- kernel 2: per-batch 64-bit radix select of K-th largest key ----------
__global__ __launch_bounds__(256) void k_radix(const float* __restrict__ scores,
                                               unsigned long long* __restrict__ thresh)
{
    const int b = blockIdx.x;
    const int tid = threadIdx.x;
    const float* sb = scores + (size_t)b * NN;
    __shared__ unsigned int hist[256];
    __shared__ unsigned long long prefS;
    __shared__ unsigned int remS;
    if (tid == 0) { prefS = 0ull; remS = KK; }
    __syncthreads();
    for (int round = 0; round < 8; ++round) {
        hist[tid] = 0u;
        __syncthreads();
        unsigned long long pref = prefS;
        int shift = 56 - 8 * round;
        for (int n = tid; n < NN; n += 256) {
            unsigned long long k = make_key(sb[n], n);
            bool match = (round == 0) || (((k ^ pref) >> (shift + 8)) == 0ull);
            if (match) atomicAdd(&hist[(unsigned int)((k >> shift) & 0xFFull)], 1u);
        }
        __syncthreads();
        if (tid == 0) {
            unsigned int rem = remS, cum = 0; int sel = 0;
            for (int bin = 255; bin >= 0; --bin) {
                unsigned int h = hist[bin];
                if (cum + h >= rem) { sel = bin; break; }
                cum += h;
            }
            remS = rem - cum;
            prefS = prefS | ((unsigned long long)sel << shift);
        }
        __syncthreads();
    }
    if (tid == 0) thresh[b] = prefS;
}

// ---------- kernel 3: compact + bitonic sort (desc) + decode pred boxes ----------
__global__ __launch_bounds__(1024) void k_topk(
    const float* __restrict__ scores, const unsigned long long* __restrict__ thresh,
    const float* __restrict__ b2d, const float* __restrict__ rois,
    const float* __restrict__ means, const float* __restrict__ stds,
    float* __restrict__ sS, float4* __restrict__ pred)
{
    const int b = blockIdx.x;
    const int tid = threadIdx.x;
    __shared__ unsigned long long keys[KK];
    __shared__ unsigned int cnt;
    keys[tid] = 0ull;
    if (tid == 0) cnt = 0u;
    __syncthreads();
    unsigned long long T = thresh[b];
    const float* sb = scores + (size_t)b * NN;
    for (int n = tid; n < NN; n += 1024) {
        unsigned long long k = make_key(sb[n], n);
        if (k >= T) {
            unsigned int pos = atomicAdd(&cnt, 1u);
            if (pos < KK) keys[pos] = k;
        }
    }
    __syncthreads();
    // bitonic sort, descending (keys unique -> deterministic)
    for (int size = 2; size <= KK; size <<= 1) {
        for (int stride = size >> 1; stride > 0; stride >>= 1) {
            int partner = tid ^ stride;
            if (partner > tid) {
                unsigned long long x = keys[tid], y = keys[partner];
                bool ascRegion = (tid & size) != 0;
                bool doSwap = ascRegion ? (x > y) : (x < y);
                if (doSwap) { keys[tid] = y; keys[partner] = x; }
            }
            __syncthreads();
        }
    }
    unsigned long long kk = keys[tid];
    unsigned int n = 0xFFFFFFFFu - (unsigned int)(kk & 0xFFFFFFFFull);
    if (n >= NN) n = 0;
    sS[(size_t)b * KK + tid] = sb[n];
    size_t i4 = ((size_t)b * NN + n) * 4;
    float d0 = b2d[i4] * stds[0] + means[0];
    float d1 = b2d[i4 + 1] * stds[1] + means[1];
    float d2 = b2d[i4 + 2] * stds[2] + means[2];
    float d3 = b2d[i4 + 3] * stds[3] + means[3];
    float4 r = ((const float4*)rois)[n];
    float aw = r.z - r.x, ah = r.w - r.y;
    float pcx = r.x + 0.5f * aw + d0 * aw;
    float pcy = r.y + 0.5f * ah + d1 * ah;
    float pw = aw * expf(fminf(fmaxf(d2, -4.f), 4.f));
    float ph = ah * expf(fminf(fmaxf(d3, -4.f), 4.f));
    pred[(size_t)b * KK + tid] = make_float4(pcx - 0.5f * pw, pcy - 0.5f * ph,
                                             pcx + 0.5f * pw, pcy + 0.5f * ph);
}

// ---------- kernel 4: blocked forward substitution (I+L)a=s, WMMA panel updates ----------
__global__ __launch_bounds__(256) void k_solve(const float4* __restrict__ pred,
                                               const float* __restrict__ sS,
                                               float* __restrict__ aOut)
{
    const int b = blockIdx.x;
    const int tid = threadIdx.x;
    const int lane = tid & 31;
    const int w = tid >> 5;               // wave id 0..7
    __shared__ float4 predL[KK];          // 16 KB: boxes resident in LDS
    __shared__ float aL[KK];
    __shared__ float sL[KK];
    __shared__ float deltaL[8][16];
    for (int i = tid; i < KK; i += 256) {
        predL[i] = pred[(size_t)b * KK + i];
        sL[i] = sS[(size_t)b * KK + i];
    }
    __syncthreads();

    for (int bi = 0; bi < KK / 16; ++bi) {
        v8f acc = {0.f, 0.f, 0.f, 0.f, 0.f, 0.f, 0.f, 0.f};
        const int m = lane & 15;
        const int kb = (lane >> 4) * 2;   // A layout: lanes 0-15 hold K=0,1; lanes 16-31 hold K=2,3
        const int row = bi * 16 + m;
        float4 pr = predL[row];
        // panel: delta = sum_{bj<bi} L(bi,bj) @ a[bj], distributed round-robin over 8 waves
        for (int bj = w; bj < bi; bj += 8) {
            #pragma unroll
            for (int k4 = 0; k4 < 4; ++k4) {
                int c0 = bj * 16 + k4 * 4 + kb;
                v2f A, Bv;
                A.x = lval(pr, predL[c0]);        // L(row, c0)   strict lower guaranteed
                A.y = lval(pr, predL[c0 + 1]);
                Bv.x = aL[c0];                    // a broadcast into all 16 columns of B
                Bv.y = aL[c0 + 1];
                acc = __builtin_amdgcn_wmma_f32_16x16x4_f32(
                    false, A, false, Bv, (short)0, acc, false, false);
            }
        }
        // extract column (all columns equal): lane0 -> M=0..7, lane16 -> M=8..15
        if (lane == 0) {
            #pragma unroll
            for (int t = 0; t < 8; ++t) deltaL[w][t] = acc[t];
        } else if (lane == 16) {
            #pragma unroll
            for (int t = 0; t < 8; ++t) deltaL[w][8 + t] = acc[t];
        }
        __syncthreads();
        // diagonal 16x16 block: serial forward substitution (unit diagonal)
        if (tid == 0) {
            for (int r = 0; r < 16; ++r) {
                int R = bi * 16 + r;
                float d = sL[R];
                #pragma unroll
                for (int ww = 0; ww < 8; ++ww) d -= deltaL[ww][r];
                for (int c = 0; c < r; ++c)
                    d -= lval(predL[R], predL[bi * 16 + c]) * aL[bi * 16 + c];
                aL[R] = d;
            }
        }
        __syncthreads();
    }
    for (int i = tid; i < KK; i += 256)
        aOut[(size_t)b * KK + i] = fminf(fmaxf(aL[i], 0.f), 1.f);
}

// ---------- kernel 5: argmax_k IoU(pred_k, gt_g), first-index tie break ----------
__global__ __launch_bounds__(32) void k_best(const float4* __restrict__ pred,
                                             const float* __restrict__ gtb,
                                             int* __restrict__ best)
{
    const int g = blockIdx.x, b = blockIdx.y, lane = threadIdx.x;
    float4 q = ((const float4*)gtb)[b * GG + g];
    float bv = -1.f; int bk = 0;
    for (int k = lane; k < KK; k += 32) {
        float v = ioup(pred[(size_t)b * KK + k], q);
        if (v > bv) { bv = v; bk = k; }
    }
    for (int off = 16; off > 0; off >>= 1) {
        float ov = __shfl_down(bv, off, 32);
        int oi = __shfl_down(bk, off, 32);
        if (ov > bv || (ov == bv && oi < bk)) { bv = ov; bk = oi; }
    }
    if (lane == 0) best[b * GG + g] = bk;
}

// ---------- kernel 6: deterministic final reduce + BCE + combine ----------
__global__ __launch_bounds__(1024) void k_final(const float4* __restrict__ partials, int nPart,
                                                const float* __restrict__ aAll,
                                                const int* __restrict__ best,
                                                float* __restrict__ out)
{
    const int tid = threadIdx.x;
    float c = 0.f, l2 = 0.f, l3 = 0.f, fgc = 0.f;
    for (int p = tid; p < nPart; p += 1024) {
        float4 v = partials[p];
        c += v.x; l2 += v.y; l3 += v.z; fgc += v.w;
    }
    __shared__ float4 red[1024];
    red[tid] = make_float4(c, l2, l3, fgc);
    __syncthreads();
    for (int s = 512; s > 0; s >>= 1) {
        if (tid < s) {
            float4 a = red[tid], b2 = red[tid + s];
            red[tid] = make_float4(a.x + b2.x, a.y + b2.y, a.z + b2.z, a.w + b2.w);
        }
        __syncthreads();
    }
    float nacc = 0.f;
    for (int i = tid; i < BB * KK; i += 1024) {
        int b = i / KK, k = i % KK;
        float a = aAll[i];
        float t = 0.f;
        for (int g = 0; g < GG; ++g)
            if (best[b * GG + g] == k) { t = 1.f; break; }
        nacc += -(t * logf(a + 1e-6f) + (1.f - t) * logf(1.f - a + 1e-6f));
    }
    __shared__ float red2[1024];
    red2[tid] = nacc;
    __syncthreads();
    for (int s = 512; s > 0; s >>= 1) {
        if (tid < s) red2[tid] += red2[tid + s];
        __syncthreads();
    }
    if (tid == 0) {
        float4 tot = red[0];
        float clsLoss = tot.x / (float)((size_t)BB * NN);
        float nfg = fmaxf(tot.w, 1.f);
        out[0] = clsLoss + tot.y / nfg + tot.z / nfg + red2[0] / (float)(BB * KK);
    }
}

// ---------- host ----------
extern "C" void kernel_launch(void* const* d_in, const int* in_sizes, int n_in,
                              void* d_out, int out_size, void* d_ws, size_t ws_size,
                              hipStream_t stream)
{
    (void)in_sizes; (void)n_in; (void)out_size; (void)ws_size;
    const float* cls   = (const float*)d_in[0];
    const float* prob  = (const float*)d_in[1];
    const float* b2d   = (const float*)d_in[2];
    const float* b3d   = (const float*)d_in[3];
    const float* rois  = (const float*)d_in[4];
    const float* gtb   = (const float*)d_in[5];
    const int*   gtl   = (const int*)d_in[6];
    const float* gt3   = (const float*)d_in[7];
    const float* means = (const float*)d_in[8];
    const float* stds  = (const float*)d_in[9];

    char* ws = (char*)d_ws;
    size_t off = 0;
    #define WS_ALLOC(ptr, type, count) \
        type* ptr = (type*)(ws + off); \
        off = (off + sizeof(type) * (size_t)(count) + 255) & ~(size_t)255;
    WS_ALLOC(scores,   float, (size_t)BB * NN)
    WS_ALLOC(partials, float4, NPART)
    WS_ALLOC(thresh,   unsigned long long, BB)
    WS_ALLOC(sSorted,  float, BB * KK)
    WS_ALLOC(pred,     float4, BB * KK)
    WS_ALLOC(aOut,     float, BB * KK)
    WS_ALLOC(best,     int, BB * GG)
    #undef WS_ALLOC

    dim3 g1(NBLK_A, BB);
    k_anchor<<<g1, 256, 0, stream>>>(cls, prob, b2d, b3d, rois, gtb, gtl, gt3,
                                     means, stds, scores, partials);
    k_radix<<<BB, 256, 0, stream>>>(scores, thresh);
    k_topk<<<BB, 1024, 0, stream>>>(scores, thresh, b2d, rois, means, stds, sSorted, pred);
    k_solve<<<BB, 256, 0, stream>>>(pred, sSorted, aOut);
    dim3 g5(GG, BB);
    k_best<<<g5, 32, 0, stream>>>(pred, gtb, best);
    k_final<<<1, 1024, 0, stream>>>(partials, NPART, aOut, best, (float*)d_out);
}